// Model_4836133175365
// MI455X (gfx1250) — compile-verified
//
#include <hip/hip_runtime.h>

// ---------------------------------------------------------------------------
// FeaStNet mesh autoencoder for MI455X (gfx1250, wave32, WMMA, async-LDS).
// Bandwidth-bound: W_enc+W_dec = 329 MB fp32 dominate (>= ~14us at 23.3TB/s).
// GEMMs use V_WMMA_F32_16X16X4_F32 (fp32 kept for 321k-long reductions).
// LDS staging uses GLOBAL_LOAD_ASYNC_TO_LDS_B32 (ASYNCcnt).
// All LDS operand panels laid out so WMMA fragments are single ds_load_b64.
// Graph conv refactored to gather-only form (src = repeat(arange(V),8)).
// ---------------------------------------------------------------------------

typedef __attribute__((ext_vector_type(2))) float v2f;
typedef __attribute__((ext_vector_type(8))) float v8f;

#define NV   5023
#define NB   64
#define KNB  8          // neighbors per vertex
#define NM   9          // attention heads
#define CHID 64
#define NLAT 128
#define RTOT 321472     // == NB*NV == NV*CHID (coincidentally equal)

static __device__ __forceinline__ int imin(int a, int b) { return a < b ? a : b; }

// ---- async global->LDS copy (one dword per lane, arbitrary LDS scatter) ----
typedef __attribute__((address_space(1))) int* g_i32p;   // global AS
typedef __attribute__((address_space(3))) int* l_i32p;   // LDS AS

static __device__ __forceinline__ void async_g2l_b32(const float* g, float* l) {
#if defined(__has_builtin) && __has_builtin(__builtin_amdgcn_global_load_async_to_lds_b32)
  __builtin_amdgcn_global_load_async_to_lds_b32(
      (g_i32p)(unsigned long long)g, (l_i32p)(unsigned long long)l, 0, 0);
#else
  unsigned int loff = (unsigned int)(unsigned long long)l;  // low 32b = LDS addr
  asm volatile("global_load_async_to_lds_b32 %0, %1, off"
               :: "v"(loff), "v"(g) : "memory");
#endif
}

static __device__ __forceinline__ void wait_async0() {
#if defined(__has_builtin) && __has_builtin(__builtin_amdgcn_s_wait_asynccnt)
  __builtin_amdgcn_s_wait_asynccnt(0);
#else
  asm volatile("s_wait_asynccnt 0x0" ::: "memory");
#endif
}

// D = A(16x4) * B(4x16) + C(16x16), fp32.
// A: lanes 0-15 rows M=0..15 hold K={0,1}; lanes 16-31 hold K={2,3}.
// B (mirror): lanes 0-15 cols N=0..15 hold K={0,1}; lanes 16-31 K={2,3}.
// C/D: vgpr r -> (M = r + 8*(lane>=16), N = lane&15).
static __device__ __forceinline__ v8f wmma4(v2f a, v2f b, v8f c) {
  return __builtin_amdgcn_wmma_f32_16x16x4_f32(false, a, false, b, (short)0, c,
                                               false, false);
}

// ---------------------------------------------------------------------------
// K1: encoder edge kernel. One thread per (b,v): softmax attention over the
// 8 owned edges, accumulate agg[m*3+c] = sum_j att_m * x_j[c]. 28-wide rows
// (col 27 zero-padded) so the following WMMA GEMM has K divisible by 4.
// ---------------------------------------------------------------------------
__global__ __launch_bounds__(256) void k_enc_edge(
    const float* __restrict__ x, const float* __restrict__ aw,
    const float* __restrict__ u_e, const float* __restrict__ c_e,
    const int* __restrict__ dst, float* __restrict__ agg) {
  int r = blockIdx.x * blockDim.x + threadIdx.x;
  if (r >= RTOT) return;
  int b = r / NV, v = r - b * NV;

  float xi[3];
#pragma unroll
  for (int c = 0; c < 3; ++c) xi[c] = x[(size_t)r * 3 + c];
  float u[27], cc[NM];
#pragma unroll
  for (int i = 0; i < 27; ++i) u[i] = u_e[i];      // u_e[c*9+m]
#pragma unroll
  for (int m = 0; m < NM; ++m) cc[m] = c_e[m];

  float acc[27];
#pragma unroll
  for (int i = 0; i < 27; ++i) acc[i] = 0.f;

  for (int k = 0; k < KNB; ++k) {
    int e = v * KNB + k;
    int j = dst[e];
    const float* xjp = x + (size_t)(b * NV + j) * 3;
    float xj0 = xjp[0], xj1 = xjp[1], xj2 = xjp[2];
    float d0 = xi[0] - xj0, d1 = xi[1] - xj1, d2 = xi[2] - xj2;
    float lg[NM], mx = -1e30f;
#pragma unroll
    for (int m = 0; m < NM; ++m) {
      lg[m] = d0 * u[0 * NM + m] + d1 * u[1 * NM + m] + d2 * u[2 * NM + m] + cc[m];
      mx = fmaxf(mx, lg[m]);
    }
    float s = 0.f;
#pragma unroll
    for (int m = 0; m < NM; ++m) { lg[m] = __expf(lg[m] - mx); s += lg[m]; }
    float sc = aw[e] / s;
#pragma unroll
    for (int m = 0; m < NM; ++m) {
      float w = lg[m] * sc;
      acc[m * 3 + 0] += w * xj0;
      acc[m * 3 + 1] += w * xj1;
      acc[m * 3 + 2] += w * xj2;
    }
  }
  float* o = agg + (size_t)r * 28;
#pragma unroll
  for (int i = 0; i < 27; ++i) o[i] = acc[i];
  o[27] = 0.f;
}

// ---------------------------------------------------------------------------
// K2: h = relu(agg[R,28] @ W_e[28(27),64] + b_e). 8 waves/block, each wave a
// 16-row tile x full 64 cols (4 accumulators). W_e async-staged into LDS
// TRANSPOSED [n][k] (stride 30) so each B fragment is one ds_load_b64.
// ---------------------------------------------------------------------------
#define WSTR2 30
__global__ __launch_bounds__(256) void k_gemm_enc_conv(
    const float* __restrict__ agg, const float* __restrict__ W_e,
    const float* __restrict__ b_e, float* __restrict__ h) {
  __shared__ float ldsWt[64 * WSTR2];   // [n][k], padded even stride
  __shared__ float ldsB[64];
  int t = threadIdx.x;
  for (int i = t; i < 28 * 64; i += 256) {
    int k = i >> 6, n = i & 63;         // W_e[k*64+n], global-coalesced in i
    if (k < 27) async_g2l_b32(W_e + i, &ldsWt[n * WSTR2 + k]);
    else        ldsWt[n * WSTR2 + k] = 0.f;
  }
  if (t < 64) ldsB[t] = b_e[t];
  wait_async0();
  __syncthreads();

  int wave = t >> 5, lane = t & 31;
  int ml = lane & 15, koff = (lane >> 4) * 2;
  int row0 = (blockIdx.x * 8 + wave) * 16;
  int arow = imin(row0 + ml, RTOT - 1);
  const float* ap = agg + (size_t)arow * 28;

  v8f acc[4];
#pragma unroll
  for (int nt = 0; nt < 4; ++nt)
#pragma unroll
    for (int q = 0; q < 8; ++q) acc[nt][q] = 0.f;

#pragma unroll
  for (int kk = 0; kk < 28; kk += 4) {
    v2f a = *(const v2f*)(ap + kk + koff);   // 8B aligned (even offsets)
#pragma unroll
    for (int nt = 0; nt < 4; ++nt) {
      v2f bf = *(const v2f*)(&ldsWt[(nt * 16 + ml) * WSTR2 + kk + koff]);
      acc[nt] = wmma4(a, bf, acc[nt]);
    }
  }
  int mbase = row0 + 8 * (lane >> 4);
#pragma unroll
  for (int nt = 0; nt < 4; ++nt) {
    int n = nt * 16 + ml;
    float bias = ldsB[n];
#pragma unroll
    for (int rr = 0; rr < 8; ++rr) {
      int m = mbase + rr;
      if (m < RTOT) {
        float val = acc[nt][rr] + bias;
        h[(size_t)m * 64 + n] = val > 0.f ? val : 0.f;
      }
    }
  }
}

// K3: z init with bias (split-K atomics accumulate on top).
__global__ void k_init_z(const float* __restrict__ b_enc, float* __restrict__ z) {
  int t = blockIdx.x * blockDim.x + threadIdx.x;
  if (t < NB * NLAT) z[t] = b_enc[t & (NLAT - 1)];
}

// ---------------------------------------------------------------------------
// K4: z += h[64,R] @ W_enc[R,128], split-K with f32 atomics.
// Block owns a 224-wide K chunk; the 64x224 h panel is async-copied into LDS
// row-major [m][k] with padded stride 226, so each A fragment is one 8B-
// aligned ds_load_b64. h read once, W_enc streamed once.
// ---------------------------------------------------------------------------
#define KCHUNK 224
#define ASTR   226
__global__ __launch_bounds__(256) void k_gemm_enc(
    const float* __restrict__ h, const float* __restrict__ W_enc,
    float* __restrict__ z) {
  __shared__ float ldsA[64 * ASTR];   // [b][k], padded even stride
  int t = threadIdx.x;
  int k0 = blockIdx.x * KCHUNK;
#pragma unroll 1
  for (int i = 0; i < (64 * KCHUNK) / 256; ++i) {  // 56 iters, coalesced
    int flat = i * 256 + t;
    int b = flat / KCHUNK;
    int k = flat - b * KCHUNK;
    int gk = k0 + k;
    if (gk < RTOT) async_g2l_b32(h + (size_t)b * RTOT + gk, &ldsA[b * ASTR + k]);
    else           ldsA[b * ASTR + k] = 0.f;       // zero pad (tail block only)
  }
  wait_async0();
  __syncthreads();

  int wave = t >> 5, lane = t & 31;
  int nl = lane & 15, koff = (lane >> 4) * 2;
  int n = wave * 16 + nl;               // N=128 covered by 8 waves

  v8f acc[4];
#pragma unroll
  for (int mt = 0; mt < 4; ++mt)
#pragma unroll
    for (int q = 0; q < 8; ++q) acc[mt][q] = 0.f;

#pragma unroll 2
  for (int kk = 0; kk < KCHUNK; kk += 4) {
    int gk = k0 + kk + koff;
    v2f bf;                              // A is zero-padded, so clamp is safe
    bf.x = W_enc[(size_t)imin(gk, RTOT - 1) * NLAT + n];
    bf.y = W_enc[(size_t)imin(gk + 1, RTOT - 1) * NLAT + n];
#pragma unroll
    for (int mt = 0; mt < 4; ++mt) {
      v2f a = *(const v2f*)(&ldsA[(mt * 16 + nl) * ASTR + kk + koff]);
      acc[mt] = wmma4(a, bf, acc[mt]);
    }
  }
#pragma unroll
  for (int mt = 0; mt < 4; ++mt)
#pragma unroll
    for (int rr = 0; rr < 8; ++rr) {
      int m = mt * 16 + rr + 8 * (lane >> 4);
      atomicAdd(&z[m * NLAT + n], acc[mt][rr]);
    }
}

// ---------------------------------------------------------------------------
// K5: d = z[64,128] @ W_dec[128,R] + b_dec. z async-staged in LDS row-major
// [m][k] (stride 130); each wave produces a 64x16 tile; block covers 128 cols.
// ---------------------------------------------------------------------------
#define ZSTR 130
__global__ __launch_bounds__(256) void k_gemm_dec(
    const float* __restrict__ z, const float* __restrict__ W_dec,
    const float* __restrict__ b_dec, float* __restrict__ d) {
  __shared__ float ldsZ[64 * ZSTR];     // [b][k], padded even stride
  int t = threadIdx.x;
#pragma unroll 1
  for (int i = 0; i < 32; ++i) {
    int flat = i * 256 + t;
    int b = flat >> 7, k = flat & 127;
    async_g2l_b32(z + flat, &ldsZ[b * ZSTR + k]);
  }
  wait_async0();
  __syncthreads();

  int wave = t >> 5, lane = t & 31;
  int nl = lane & 15, koff = (lane >> 4) * 2;
  int n = blockIdx.x * 128 + wave * 16 + nl;
  int nc = imin(n, RTOT - 1);

  v8f acc[4];
#pragma unroll
  for (int mt = 0; mt < 4; ++mt)
#pragma unroll
    for (int q = 0; q < 8; ++q) acc[mt][q] = 0.f;

#pragma unroll 2
  for (int kk = 0; kk < NLAT; kk += 4) {
    v2f bf;
    bf.x = W_dec[(size_t)(kk + koff) * RTOT + nc];
    bf.y = W_dec[(size_t)(kk + koff + 1) * RTOT + nc];
#pragma unroll
    for (int mt = 0; mt < 4; ++mt) {
      v2f a = *(const v2f*)(&ldsZ[(mt * 16 + nl) * ZSTR + kk + koff]);
      acc[mt] = wmma4(a, bf, acc[mt]);
    }
  }
  if (n < RTOT) {
    float bias = b_dec[n];
#pragma unroll
    for (int mt = 0; mt < 4; ++mt)
#pragma unroll
      for (int rr = 0; rr < 8; ++rr) {
        int m = mt * 16 + rr + 8 * (lane >> 4);
        d[(size_t)m * RTOT + n] = acc[mt][rr] + bias;
      }
  }
}

// ---------------------------------------------------------------------------
// K6: fused decoder weight [64 x 48]: cols 0..26 = W_d[m][c][o] at (m*3+o),
// cols 27..35 = u_d[c][m], cols 36..47 = 0 (pad to 3 N-tiles).
// ---------------------------------------------------------------------------
__global__ void k_prep_wcomb(const float* __restrict__ W_d,
                             const float* __restrict__ u_d,
                             float* __restrict__ Wc) {
  int i = blockIdx.x * blockDim.x + threadIdx.x;
  if (i >= 64 * 48) return;
  int c = i / 48, col = i - c * 48;
  float val = 0.f;
  if (col < 27) {
    int m = col / 3, o = col - m * 3;
    val = W_d[m * 192 + c * 3 + o];     // W_d[9][64][3]
  } else if (col < 36) {
    val = u_d[c * NM + (col - 27)];     // u_d[64][9]
  }
  Wc[i] = val;
}

// ---------------------------------------------------------------------------
// K7: tdp[R,48] = d[R,64] @ Wcomb[64,48]  (t_j transforms + attention logits
// p = d@u_d in one pass). Weight async-staged into LDS TRANSPOSED [n][k]
// (stride 66) so each B fragment is one ds_load_b64; 16-row tiles per wave.
// ---------------------------------------------------------------------------
#define WSTR7 66
__global__ __launch_bounds__(256) void k_gemm_tdp(
    const float* __restrict__ d, const float* __restrict__ Wc,
    float* __restrict__ tdp) {
  __shared__ float ldsWt[48 * WSTR7];   // [n][k], padded even stride
  int t = threadIdx.x;
  for (int i = t; i < 64 * 48; i += 256) {
    int k = i / 48, n = i - k * 48;     // Wc[k*48+n], global-coalesced in i
    async_g2l_b32(Wc + i, &ldsWt[n * WSTR7 + k]);
  }
  wait_async0();
  __syncthreads();

  int wave = t >> 5, lane = t & 31;
  int ml = lane & 15, koff = (lane >> 4) * 2;
  int row0 = (blockIdx.x * 8 + wave) * 16;
  int ar = imin(row0 + ml, RTOT - 1);
  const float* ap = d + (size_t)ar * 64;

  v8f acc[3];
#pragma unroll
  for (int nt = 0; nt < 3; ++nt)
#pragma unroll
    for (int q = 0; q < 8; ++q) acc[nt][q] = 0.f;

#pragma unroll
  for (int kk = 0; kk < 64; kk += 4) {
    v2f a = *(const v2f*)(ap + kk + koff);
#pragma unroll
    for (int nt = 0; nt < 3; ++nt) {
      v2f bf = *(const v2f*)(&ldsWt[(nt * 16 + ml) * WSTR7 + kk + koff]);
      acc[nt] = wmma4(a, bf, acc[nt]);
    }
  }
  int mbase = row0 + 8 * (lane >> 4);
#pragma unroll
  for (int nt = 0; nt < 3; ++nt)
#pragma unroll
    for (int rr = 0; rr < 8; ++rr) {
      int m = mbase + rr;
      if (m < RTOT) tdp[(size_t)m * 48 + nt * 16 + ml] = acc[nt][rr];
    }
}

// ---------------------------------------------------------------------------
// K8: decoder edge kernel. Gather 36 floats per neighbor (t_j[27] + p_j[9],
// 9x float4, 192B row stride), softmax over heads, 3 outputs + relu.
// Working set (62MB) fits in the 192MB L2 -> gathers stay on-die.
// ---------------------------------------------------------------------------
__global__ __launch_bounds__(256) void k_dec_edge(
    const float* __restrict__ tdp, const float* __restrict__ aw,
    const float* __restrict__ c_d, const float* __restrict__ b_d,
    const int* __restrict__ dst, float* __restrict__ out) {
  int r = blockIdx.x * blockDim.x + threadIdx.x;
  if (r >= RTOT) return;
  int b = r / NV, v = r - b * NV;

  float pi[NM], cc[NM];
  const float* rp = tdp + (size_t)r * 48 + 27;
#pragma unroll
  for (int m = 0; m < NM; ++m) { pi[m] = rp[m]; cc[m] = c_d[m]; }

  float a0 = 0.f, a1 = 0.f, a2 = 0.f;
  for (int k = 0; k < KNB; ++k) {
    int e = v * KNB + k;
    int j = dst[e];
    const float4* q4 = (const float4*)(tdp + (size_t)(b * NV + j) * 48);
    float tj[36];
#pragma unroll
    for (int i = 0; i < 9; ++i) {
      float4 f = q4[i];
      tj[i * 4 + 0] = f.x; tj[i * 4 + 1] = f.y;
      tj[i * 4 + 2] = f.z; tj[i * 4 + 3] = f.w;
    }
    float lg[NM], mx = -1e30f;
#pragma unroll
    for (int m = 0; m < NM; ++m) {
      lg[m] = pi[m] - tj[27 + m] + cc[m];
      mx = fmaxf(mx, lg[m]);
    }
    float s = 0.f;
#pragma unroll
    for (int m = 0; m < NM; ++m) { lg[m] = __expf(lg[m] - mx); s += lg[m]; }
    float sc = aw[e] / s;
#pragma unroll
    for (int m = 0; m < NM; ++m) {
      float w = lg[m] * sc;
      a0 += w * tj[m * 3 + 0];
      a1 += w * tj[m * 3 + 1];
      a2 += w * tj[m * 3 + 2];
    }
  }
  a0 += b_d[0]; a1 += b_d[1]; a2 += b_d[2];
  out[(size_t)r * 3 + 0] = a0 > 0.f ? a0 : 0.f;
  out[(size_t)r * 3 + 1] = a1 > 0.f ? a1 : 0.f;
  out[(size_t)r * 3 + 2] = a2 > 0.f ? a2 : 0.f;
}

// ---------------------------------------------------------------------------
extern "C" void kernel_launch(void* const* d_in, const int* in_sizes, int n_in,
                              void* d_out, int out_size, void* d_ws,
                              size_t ws_size, hipStream_t stream) {
  const float* x     = (const float*)d_in[0];
  const float* aw    = (const float*)d_in[1];
  const float* u_e   = (const float*)d_in[2];
  const float* c_e   = (const float*)d_in[3];
  const float* W_e   = (const float*)d_in[4];
  const float* b_e   = (const float*)d_in[5];
  const float* W_enc = (const float*)d_in[6];
  const float* b_enc = (const float*)d_in[7];
  const float* W_dec = (const float*)d_in[8];
  const float* b_dec = (const float*)d_in[9];
  const float* u_d   = (const float*)d_in[10];
  const float* c_d   = (const float*)d_in[11];
  const float* W_d   = (const float*)d_in[12];
  const float* b_d   = (const float*)d_in[13];
  const int*   dst   = (const int*)d_in[15];   // src = repeat(arange(V),8) by construction
  float* out = (float*)d_out;

  // workspace layout (bytes), tdp reuses h region (h dead after K4)
  char* ws = (char*)d_ws;
  float* agg = (float*)(ws);                             // R*28*4 = 36,004,864
  float* h   = (float*)(ws + 36004864);                  // R*64*4 = 82,296,832
  float* d   = (float*)(ws + 36004864 + 82296832);       // R*64*4 = 82,296,832
  float* z   = (float*)(ws + 36004864 + 82296832 + 82296832);        // 32,768
  float* Wc  = (float*)(ws + 36004864 + 82296832 + 82296832 + 32768);// 12,288
  float* tdp = h;                                        // R*48*4 = 61,722,624 <= h size

  const int R = RTOT;
  dim3 blk(256);

  k_enc_edge<<<dim3((R + 255) / 256), blk, 0, stream>>>(x, aw, u_e, c_e, dst, agg);
  k_gemm_enc_conv<<<dim3((R + 127) / 128), blk, 0, stream>>>(agg, W_e, b_e, h);
  k_init_z<<<dim3((NB * NLAT + 255) / 256), blk, 0, stream>>>(b_enc, z);
  k_gemm_enc<<<dim3((R + KCHUNK - 1) / KCHUNK), blk, 0, stream>>>(h, W_enc, z);
  k_gemm_dec<<<dim3((R + 127) / 128), blk, 0, stream>>>(z, W_dec, b_dec, d);
  k_prep_wcomb<<<dim3((64 * 48 + 255) / 256), blk, 0, stream>>>(W_d, u_d, Wc);
  k_gemm_tdp<<<dim3((R + 127) / 128), blk, 0, stream>>>(d, Wc, tdp);
  k_dec_edge<<<dim3((R + 255) / 256), blk, 0, stream>>>(tdp, aw, c_d, b_d, dst, out);
}